// ChamferLoss_58652073394562
// MI455X (gfx1250) — compile-verified
//
#include <hip/hip_runtime.h>
#include <math.h>

typedef __attribute__((ext_vector_type(2))) float v2f;
typedef __attribute__((ext_vector_type(8))) float v8f;

#define KPTS 8192
#define BATCH 4
#define TILE 16
#define WAVES_PB 8
#define ROWS_PW 32                      // 2 i-tiles per wave
#define ROWS_PB (WAVES_PB * ROWS_PW)    // 256 pred rows per block
#define JTILES (KPTS / TILE)            // 512
#define INF_BITS 0x7F800000u

// ---------------------------------------------------------------------------
// init: fill col-min buffer with +inf bit pattern
// ---------------------------------------------------------------------------
__global__ void chamfer_init(unsigned* __restrict__ buf, int n) {
    int i = blockIdx.x * blockDim.x + threadIdx.x;
    if (i < n) buf[i] = INF_BITS;
}

// ---------------------------------------------------------------------------
// main: per-block strip of 256 pred rows; each wave owns TWO 16-row i-tiles
// and sweeps target tiles TWO j-tiles per iteration (4 WMMAs/iter).
//   A (16x4)  : lane l -> M = l&15, K = (l>>4)*2 + {0,1}; pre-scaled by -2
//   B (4x16)  : lane l -> N = l&15, K = (l>>4)*2 + {0,1}
//   C (16x16) : loop-invariant |p_i|^2 (row-indexed == D layout)
//   D[v]      : p2[i] - 2*dot(i,j);  d^2 = D[v] + t2[j]
// t2-adds done as v8f + scalar (in-tile adjacent pk_add, broadcast op_sel),
// min work via v_min3 on arbitrary regs -> no cross-tile register shuffling.
// ---------------------------------------------------------------------------
__global__ __launch_bounds__(256)
void chamfer_tiles(const float* __restrict__ pred,
                   const float* __restrict__ target,
                   float* __restrict__ rowmin,      // B*K, d^2 (clamped in reduce)
                   unsigned* __restrict__ colmin) { // B*K, d^2 bits (>= 0)
    __shared__ unsigned ldsCol[KPTS];               // 32 KB block-local col mins

    const int tid  = threadIdx.x;
    const int lane = tid & 31;
    const int wave = tid >> 5;
    const int b    = blockIdx.x / (KPTS / ROWS_PB);
    const int strip = (blockIdx.x % (KPTS / ROWS_PB)) * ROWS_PB;

    for (int j = tid; j < KPTS; j += 256) ldsCol[j] = INF_BITS;
    __syncthreads();

    const float* Pb = pred   + (size_t)b * KPTS * 3;
    const float* Tb = target + (size_t)b * KPTS * 3;

    const int hi = lane >> 4;     // 0: K=0,1  1: K=2,3
    const int lo = lane & 15;

    // Two A matrices (rows arow0, arow0+16), scaled by -2, loop-invariant
    const int arow0 = strip + wave * ROWS_PW + lo;
    v2f a0, a1;
    a0.x = -2.0f * Pb[(arow0)      * 3 + (hi ? 2 : 0)];
    a0.y = hi ? 0.0f : (-2.0f * Pb[(arow0) * 3 + 1]);
    a1.x = -2.0f * Pb[(arow0 + 16) * 3 + (hi ? 2 : 0)];
    a1.y = hi ? 0.0f : (-2.0f * Pb[(arow0 + 16) * 3 + 1]);

    // Two C matrices = |p|^2 per D-row, loop-invariant
    v8f cp2a, cp2b;
#pragma unroll
    for (int v = 0; v < 8; ++v) {
        int r0 = strip + wave * ROWS_PW + v + 8 * hi;
        float x = Pb[r0 * 3 + 0], y = Pb[r0 * 3 + 1], z = Pb[r0 * 3 + 2];
        cp2a[v] = x * x + y * y + z * z;
        int r1 = r0 + 16;
        x = Pb[r1 * 3 + 0]; y = Pb[r1 * 3 + 1]; z = Pb[r1 * 3 + 2];
        cp2b[v] = x * x + y * y + z * z;
    }

    float rowMinA[8], rowMinB[8];
#pragma unroll
    for (int v = 0; v < 8; ++v) {
        rowMinA[v] = __uint_as_float(INF_BITS);
        rowMinB[v] = __uint_as_float(INF_BITS);
    }

    // process a PAIR of j-tiles (jt, jt+1)
    auto processPair = [&](int jt,
                           float tx0, float ty0, float tz0,
                           float tx1, float ty1, float tz1) {
        float t2_0 = tx0 * tx0 + ty0 * ty0 + tz0 * tz0;
        float t2_1 = tx1 * tx1 + ty1 * ty1 + tz1 * tz1;
        v2f bm0, bm1;
        bm0.x = hi ? tz0 : tx0;  bm0.y = hi ? 0.0f : ty0;
        bm1.x = hi ? tz1 : tx1;  bm1.y = hi ? 0.0f : ty1;

        v8f dA0 = __builtin_amdgcn_wmma_f32_16x16x4_f32(
            false, a0, false, bm0, (short)0, cp2a, false, false);
        v8f dB0 = __builtin_amdgcn_wmma_f32_16x16x4_f32(
            false, a1, false, bm0, (short)0, cp2b, false, false);
        v8f dA1 = __builtin_amdgcn_wmma_f32_16x16x4_f32(
            false, a0, false, bm1, (short)0, cp2a, false, false);
        v8f dB1 = __builtin_amdgcn_wmma_f32_16x16x4_f32(
            false, a1, false, bm1, (short)0, cp2b, false, false);

        // whole-vector adds: adjacent in-tile pk_add with scalar broadcast
        v8f eA0 = dA0 + t2_0;
        v8f eB0 = dB0 + t2_0;
        v8f eA1 = dA1 + t2_1;
        v8f eB1 = dB1 + t2_1;

        // row mins: min3(rowMin, e_jt0, e_jt1)
#pragma unroll
        for (int v = 0; v < 8; ++v)
            rowMinA[v] = fminf(rowMinA[v], fminf(eA0[v], eA1[v]));
#pragma unroll
        for (int v = 0; v < 8; ++v)
            rowMinB[v] = fminf(rowMinB[v], fminf(eB0[v], eB1[v]));

        // column mins: per-j-tile tree over this lane's 16 rows
        float cmin0 = eA0[0];
        float cmin1 = eA1[0];
#pragma unroll
        for (int v = 1; v < 8; ++v) {
            cmin0 = fminf(cmin0, eA0[v]);
            cmin1 = fminf(cmin1, eA1[v]);
        }
#pragma unroll
        for (int v = 0; v < 8; ++v) {
            cmin0 = fminf(cmin0, eB0[v]);
            cmin1 = fminf(cmin1, eB1[v]);
        }
        atomicMin(&ldsCol[jt * TILE + lo],
                  __float_as_uint(fmaxf(cmin0, 0.0f)));
        atomicMin(&ldsCol[(jt + 1) * TILE + lo],
                  __float_as_uint(fmaxf(cmin1, 0.0f)));
    };

    // software pipeline, 2 tiles deep, peeled tail (no wrap address math)
    const float* tp = Tb + (size_t)lo * 3;
    float tx0 = tp[0], ty0 = tp[1], tz0 = tp[2];
    float tx1 = tp[TILE * 3 + 0], ty1 = tp[TILE * 3 + 1], tz1 = tp[TILE * 3 + 2];

    for (int jt = 0; jt < JTILES - 2; jt += 2) {
        const float* np0 = tp + (size_t)(jt + 2) * TILE * 3;
        const float* np1 = tp + (size_t)(jt + 3) * TILE * 3;
        float nx0 = np0[0], ny0 = np0[1], nz0 = np0[2];
        float nx1 = np1[0], ny1 = np1[1], nz1 = np1[2];
        processPair(jt, tx0, ty0, tz0, tx1, ty1, tz1);
        tx0 = nx0; ty0 = ny0; tz0 = nz0;
        tx1 = nx1; ty1 = ny1; tz1 = nz1;
    }
    processPair(JTILES - 2, tx0, ty0, tz0, tx1, ty1, tz1);

    // row mins: reduce across the 16 lanes of each half-wave
#pragma unroll
    for (int v = 0; v < 8; ++v) {
        float r0 = rowMinA[v];
        float r1 = rowMinB[v];
#pragma unroll
        for (int m = 1; m < 16; m <<= 1) {
            r0 = fminf(r0, __shfl_xor(r0, m, 32));
            r1 = fminf(r1, __shfl_xor(r1, m, 32));
        }
        if (lo == 0) {
            int row = strip + wave * ROWS_PW + v + 8 * hi;
            rowmin[(size_t)b * KPTS + row]      = r0;
            rowmin[(size_t)b * KPTS + row + 16] = r1;
        }
    }

    __syncthreads();
    // merge block-local column mins into global buffer
    for (int j = tid; j < KPTS; j += 256) {
        atomicMin(&colmin[(size_t)b * KPTS + j], ldsCol[j]);
    }
}

// ---------------------------------------------------------------------------
// final: sqrt + masked sums + scalar loss (single block)
// ---------------------------------------------------------------------------
__global__ __launch_bounds__(256)
void chamfer_reduce(const float* __restrict__ rowmin,
                    const unsigned* __restrict__ colmin,
                    const float* __restrict__ mask,
                    float* __restrict__ out) {
    __shared__ float s1[256], s2[256], sm[256];
    float a = 0.0f, c = 0.0f, m = 0.0f;
    for (int i = threadIdx.x; i < BATCH * KPTS; i += 256) {
        float mk = mask[i];
        a += sqrtf(fmaxf(rowmin[i], 0.0f)) * mk;
        c += sqrtf(fmaxf(__uint_as_float(colmin[i]), 0.0f)) * mk;
        m += mk;
    }
    s1[threadIdx.x] = a; s2[threadIdx.x] = c; sm[threadIdx.x] = m;
    __syncthreads();
    for (int s = 128; s > 0; s >>= 1) {
        if (threadIdx.x < s) {
            s1[threadIdx.x] += s1[threadIdx.x + s];
            s2[threadIdx.x] += s2[threadIdx.x + s];
            sm[threadIdx.x] += sm[threadIdx.x + s];
        }
        __syncthreads();
    }
    if (threadIdx.x == 0)
        out[0] = (s1[0] + s2[0]) / (2.0f * (sm[0] + 1e-8f));
}

// ---------------------------------------------------------------------------
extern "C" void kernel_launch(void* const* d_in, const int* in_sizes, int n_in,
                              void* d_out, int out_size, void* d_ws, size_t ws_size,
                              hipStream_t stream) {
    const float* pred   = (const float*)d_in[0];   // (B,K,3) f32
    const float* target = (const float*)d_in[1];   // (B,K,3) f32
    const float* mask   = (const float*)d_in[2];   // (B,K)   f32
    float*       out    = (float*)d_out;           // scalar f32

    const int NBK = BATCH * KPTS;                  // 32768
    float*    rowmin = (float*)d_ws;               // 128 KB
    unsigned* colmin = (unsigned*)((char*)d_ws + (size_t)NBK * sizeof(float));

    chamfer_init<<<(NBK + 255) / 256, 256, 0, stream>>>(colmin, NBK);

    const int nblocks = BATCH * (KPTS / ROWS_PB);  // 128
    chamfer_tiles<<<nblocks, 256, 0, stream>>>(pred, target, rowmin, colmin);

    chamfer_reduce<<<1, 256, 0, stream>>>(rowmin, colmin, mask, out);
}